// PointwiseGraphNN_82300163326555
// MI455X (gfx1250) — compile-verified
//
#include <hip/hip_runtime.h>
#include <math.h>

// ---------------- problem constants (from reference) ----------------
#define T_    8
#define N_    2
#define C_    256
#define H1_   32
#define W1_   32
#define P_    8192          // T*H1*W1
#define Q_    4096          // H2*W2
#define CQK_  32
#define KEPT_ 4096          // P * REL_PERCENT
#define SCALE_ 0.17677669529663687f   // 1/sqrt(32)

typedef __attribute__((ext_vector_type(2))) float v2f;
typedef __attribute__((ext_vector_type(8))) float v8f;

// D(16x16) = A(16x4) * B(4x16) + C, fp32 WMMA (CDNA5, wave32)
__device__ __forceinline__ v8f wmma_f32(v2f a, v2f b, v8f c) {
  return __builtin_amdgcn_wmma_f32_16x16x4_f32(
      /*neg_a=*/false, a, /*neg_b=*/false, b,
      /*c_mod=*/(short)0, c, /*reuse_a=*/false, /*reuse_b=*/false);
}

// ---------------------------------------------------------------
// K1: projection GEMM  Y[n][o][l] = sum_c W[o][c] * X(n,c,l) + bias[o]
// X column addressing: l -> t = l/S, s = l%S; addr = n*nOff + t*Tstr + c*Cstr + s
// One wave per 16x16 output tile.
// ---------------------------------------------------------------
__global__ void proj_gemm(const float* __restrict__ W, const float* __restrict__ bias,
                          const float* __restrict__ X, float* __restrict__ Y,
                          int O, int Cdim, int Ldim, int S,
                          long Tstr, long Cstr, long nOff, long Yn) {
  const int lane = threadIdx.x & 31;
  const int wave = threadIdx.x >> 5;
  const int l16 = lane & 15, hl = lane >> 4;
  const int n = blockIdx.z;
  const int ltiles = Ldim >> 4;
  const int tiles = (O >> 4) * ltiles;
  const int tile = blockIdx.x * (blockDim.x >> 5) + wave;
  if (tile >= tiles) return;                 // whole-wave uniform: EXEC stays all-1s
  const int lt = tile % ltiles;
  const int ot = tile / ltiles;
  const float* Xn = X + (long)n * nOff;
  float* Yo = Y + (long)n * Yn;
  const int l0 = lt << 4;
  const int t = l0 / S;
  const int s = l0 - t * S;
  const float* Xcol = Xn + (long)t * Tstr + s;
  const int arow = (ot << 4) + l16;
  v8f acc = {};
  for (int k0 = 0; k0 < Cdim; k0 += 4) {
    const int ka = k0 + hl * 2;
    v2f a, b;
    a.x = W[(long)arow * Cdim + ka];
    a.y = W[(long)arow * Cdim + ka + 1];
    const float* xp = Xcol + (long)ka * Cstr;
    b.x = xp[l16];
    b.y = xp[Cstr + l16];
    acc = wmma_f32(a, b, acc);
  }
#pragma unroll
  for (int i = 0; i < 8; ++i) {
    const int orow = (ot << 4) + hl * 8 + i;
    Yo[(long)orow * Ldim + l0 + l16] = acc[i] + bias[orow];
  }
}

// ---------------------------------------------------------------
// K2: per-column (over P) online-softmax stats of rel = (qm^T km)*SCALE
// Each block: one (n, q-tile). 8 waves split the 512 p-tiles; merge
// lane<->lane+16 then across waves via LDS. Never materializes rel.
// ---------------------------------------------------------------
__global__ void col_stats(const float* __restrict__ qm, const float* __restrict__ km,
                          float* __restrict__ colmax, float* __restrict__ colden) {
  __shared__ float mSh[8][16];
  __shared__ float sSh[8][16];
  const int lane = threadIdx.x & 31, wave = threadIdx.x >> 5;
  const int l16 = lane & 15, hl = lane >> 4;
  const int n = blockIdx.z, qt = blockIdx.x;
  const float* qmN = qm + (long)n * CQK_ * P_;
  const float* kmN = km + (long)n * CQK_ * Q_;
  float m = -INFINITY, s = 0.f;
  for (int pt = wave; pt < (P_ >> 4); pt += 8) {
    v8f acc = {};
#pragma unroll
    for (int ko = 0; ko < 8; ++ko) {
      const int ka = ko * 4 + hl * 2;
      v2f a, b;
      a.x = qmN[(long)ka * P_ + (pt << 4) + l16];
      a.y = qmN[(long)(ka + 1) * P_ + (pt << 4) + l16];
      b.x = kmN[(long)ka * Q_ + (qt << 4) + l16];
      b.y = kmN[(long)(ka + 1) * Q_ + (qt << 4) + l16];
      acc = wmma_f32(a, b, acc);
    }
#pragma unroll
    for (int i = 0; i < 8; ++i) {
      const float x = acc[i] * SCALE_;
      const float M = fmaxf(m, x);
      s = s * __expf(m - M) + __expf(x - M);
      m = M;
    }
  }
  { // merge the two 16-lane halves (same q column)
    const float m2 = __shfl_xor(m, 16);
    const float s2 = __shfl_xor(s, 16);
    const float M = fmaxf(m, m2);
    s = s * __expf(m - M) + s2 * __expf(m2 - M);
    m = M;
  }
  if (lane < 16) { mSh[wave][lane] = m; sSh[wave][lane] = s; }
  __syncthreads();
  if (threadIdx.x < 16) {
    float mm = mSh[0][threadIdx.x], ss = sSh[0][threadIdx.x];
    for (int w = 1; w < 8; ++w) {
      const float mw = mSh[w][threadIdx.x], sw = sSh[w][threadIdx.x];
      const float M = fmaxf(mm, mw);
      ss = ss * __expf(mm - M) + sw * __expf(mw - M);
      mm = M;
    }
    colmax[(long)n * Q_ + (qt << 4) + threadIdx.x] = mm;
    colden[(long)n * Q_ + (qt << 4) + threadIdx.x] = ss;
  }
}

// ---------------------------------------------------------------
// K3: rel_sum[p] = sum_{n,q} softmax(rel)[n,p,q]  (rel regenerated on the fly)
// One wave per (n, p-tile); exactly 2 atomic contributions per address
// (n=0,n=1) -> commutative -> deterministic.
// ---------------------------------------------------------------
__global__ void rel_rowsum(const float* __restrict__ qm, const float* __restrict__ km,
                           const float* __restrict__ colmax, const float* __restrict__ colden,
                           float* __restrict__ relsum) {
  const int lane = threadIdx.x & 31, wave = threadIdx.x >> 5;
  const int l16 = lane & 15, hl = lane >> 4;
  const int n = blockIdx.z;
  const int pt = blockIdx.x * 8 + wave;
  const float* qmN = qm + (long)n * CQK_ * P_;
  const float* kmN = km + (long)n * CQK_ * Q_;
  float racc[8] = {};
  for (int qt = 0; qt < (Q_ >> 4); ++qt) {
    v8f acc = {};
#pragma unroll
    for (int ko = 0; ko < 8; ++ko) {
      const int ka = ko * 4 + hl * 2;
      v2f a, b;
      a.x = qmN[(long)ka * P_ + (pt << 4) + l16];
      a.y = qmN[(long)(ka + 1) * P_ + (pt << 4) + l16];
      b.x = kmN[(long)ka * Q_ + (qt << 4) + l16];
      b.y = kmN[(long)(ka + 1) * Q_ + (qt << 4) + l16];
      acc = wmma_f32(a, b, acc);
    }
    const float cm = colmax[(long)n * Q_ + (qt << 4) + l16];
    const float rd = 1.0f / colden[(long)n * Q_ + (qt << 4) + l16];
#pragma unroll
    for (int i = 0; i < 8; ++i)
      racc[i] += __expf(acc[i] * SCALE_ - cm) * rd;
  }
#pragma unroll
  for (int off = 1; off < 16; off <<= 1)
#pragma unroll
    for (int i = 0; i < 8; ++i)
      racc[i] += __shfl_xor(racc[i], off);
  if (l16 == 0) {
#pragma unroll
    for (int i = 0; i < 8; ++i)
      atomicAdd(&relsum[(pt << 4) + hl * 8 + i], racc[i]);
  }
}

// ---------------------------------------------------------------
// K4: top-KEPT selection. Bitonic-sort 8192 floats in LDS to find the
// threshold; then a deterministic stable scan builds the kept index set
// (set order is irrelevant to the final sum).
// ---------------------------------------------------------------
__global__ void select_topk(const float* __restrict__ relsum, int* __restrict__ keptIdx) {
  __shared__ float v[P_];
  const int tid = threadIdx.x;
  for (int i = tid; i < P_; i += blockDim.x) v[i] = relsum[i];
  __syncthreads();
  for (int k = 2; k <= P_; k <<= 1) {
    for (int j = k >> 1; j > 0; j >>= 1) {
      for (int i = tid; i < P_; i += blockDim.x) {
        const int ixj = i ^ j;
        if (ixj > i) {
          const bool desc = ((i & k) == 0);
          const float a = v[i], b = v[ixj];
          if (desc ? (a < b) : (a > b)) { v[i] = b; v[ixj] = a; }
        }
      }
      __syncthreads();
    }
  }
  if (tid == 0) {
    const float thr = v[KEPT_ - 1];        // KEPT-th largest
    int g = 0;
    for (int p = 0; p < P_; ++p) if (relsum[p] > thr) ++g;
    int quota = KEPT_ - g, c = 0;
    for (int p = 0; p < P_ && c < KEPT_; ++p) {
      const float x = relsum[p];
      if (x > thr) keptIdx[c++] = p;
      else if (x == thr && quota > 0) { keptIdx[c++] = p; --quota; }
    }
  }
}

// K5: column compaction  out[r][j] = in[r][idx[j]]
__global__ void compact_cols(const float* __restrict__ in, float* __restrict__ out,
                             const int* __restrict__ idx, int inStride) {
  const int j = blockIdx.x * blockDim.x + threadIdx.x;
  const int r = blockIdx.y;
  if (j < KEPT_) out[(long)r * KEPT_ + j] = in[(long)r * inStride + idx[j]];
}

// ---------------------------------------------------------------
// K6: fused output GEMM. Block = (n, 16-wide q-tile), 8 waves.
// Per 128-wide j chunk: each wave regenerates one 16x16 probability tile
// (inner K=32 WMMA + softmax from stored stats) into LDS; after barrier,
// every wave consumes all 8 tiles as the B operand of the big K=4096 GEMM,
// each wave owning 32 rows of C. out[n][c][q] = sum_j vc[c][j]*probs[j][q].
// ---------------------------------------------------------------
__global__ void out_gemm(const float* __restrict__ qc, const float* __restrict__ kmm,
                         const float* __restrict__ vc, const float* __restrict__ colmax,
                         const float* __restrict__ colden, float* __restrict__ out) {
  __shared__ float probs[8][16][16];
  const int lane = threadIdx.x & 31, wave = threadIdx.x >> 5;
  const int l16 = lane & 15, hl = lane >> 4;
  const int n = blockIdx.z, qt = blockIdx.x;
  const float* qcN = qc + (long)n * CQK_ * KEPT_;
  const float* kmN = kmm + (long)n * CQK_ * Q_;
  const float* vcN = vc + (long)n * C_ * KEPT_;
  const float cm = colmax[(long)n * Q_ + (qt << 4) + l16];
  const float rd = 1.0f / colden[(long)n * Q_ + (qt << 4) + l16];
  const int c0 = wave << 5;                  // this wave owns C rows [c0, c0+32)
  v8f acc0 = {}, acc1 = {};
  for (int jc = 0; jc < KEPT_; jc += 128) {
    // --- regenerate one probability tile per wave ---
    const int j0 = jc + (wave << 4);
    v8f pa = {};
#pragma unroll
    for (int ko = 0; ko < 8; ++ko) {
      const int ka = ko * 4 + hl * 2;
      v2f a, b;
      a.x = qcN[(long)ka * KEPT_ + j0 + l16];
      a.y = qcN[(long)(ka + 1) * KEPT_ + j0 + l16];
      b.x = kmN[(long)ka * Q_ + (qt << 4) + l16];
      b.y = kmN[(long)(ka + 1) * Q_ + (qt << 4) + l16];
      pa = wmma_f32(a, b, pa);
    }
#pragma unroll
    for (int i = 0; i < 8; ++i)
      probs[wave][hl * 8 + i][l16] = __expf(pa[i] * SCALE_ - cm) * rd;
    __syncthreads();
    // hint-prefetch next V chunk (global_prefetch_b8)
    if (jc + 128 < KEPT_)
      __builtin_prefetch(&vcN[(long)(c0 + l16) * KEPT_ + jc + 128 + hl * 64], 0, 1);
    // --- consume all 8 tiles: D += A(vc) x B(probs) ---
#pragma unroll 2
    for (int cc = 0; cc < 8; ++cc) {
#pragma unroll
      for (int kk = 0; kk < 4; ++kk) {
        const int kb = kk * 4 + hl * 2;
        v2f b, a0, a1;
        b.x = probs[cc][kb][l16];
        b.y = probs[cc][kb + 1][l16];
        const long col = jc + (cc << 4) + kb;
        const float* r0 = &vcN[(long)(c0 + l16) * KEPT_ + col];
        const float* r1 = &vcN[(long)(c0 + 16 + l16) * KEPT_ + col];
        a0.x = r0[0]; a0.y = r0[1];
        a1.x = r1[0]; a1.y = r1[1];
        acc0 = wmma_f32(a0, b, acc0);
        acc1 = wmma_f32(a1, b, acc1);
      }
    }
    __syncthreads();
  }
#pragma unroll
  for (int i = 0; i < 8; ++i) {
    const int r0 = c0 + hl * 8 + i;
    out[((long)n * C_ + r0) * Q_ + (qt << 4) + l16] = acc0[i];
    out[((long)n * C_ + r0 + 16) * Q_ + (qt << 4) + l16] = acc1[i];
  }
}

// ---------------------------------------------------------------
extern "C" void kernel_launch(void* const* d_in, const int* in_sizes, int n_in,
                              void* d_out, int out_size, void* d_ws, size_t ws_size,
                              hipStream_t stream) {
  (void)in_sizes; (void)n_in; (void)out_size;
  const float* x  = (const float*)d_in[0];   // space_time_mem (T,N,C,H1,W1)
  const float* f2 = (const float*)d_in[1];   // features2 (N,C,H2,W2)
  const float* Wq = (const float*)d_in[2];
  const float* bq = (const float*)d_in[3];
  const float* Wk = (const float*)d_in[4];
  const float* bk = (const float*)d_in[5];
  const float* Wv = (const float*)d_in[6];
  const float* bv = (const float*)d_in[7];
  float* out = (float*)d_out;

  float* ws     = (float*)d_ws;
  float* qm     = ws;                                  // [N][32][P]
  float* km     = qm + (size_t)N_ * CQK_ * P_;         // [N][32][Q]
  float* vm     = km + (size_t)N_ * CQK_ * Q_;         // [N][C][P]
  float* colmax = vm + (size_t)N_ * C_ * P_;           // [N][Q]
  float* colden = colmax + (size_t)N_ * Q_;            // [N][Q]
  float* relsum = colden + (size_t)N_ * Q_;            // [P]
  int*   kept   = (int*)(relsum + P_);                 // [KEPT]
  float* qc     = (float*)(kept + KEPT_);              // [N][32][KEPT]
  float* vc     = qc + (size_t)N_ * CQK_ * KEPT_;      // [N][C][KEPT]
  const size_t need = ((size_t)(vc - ws) + (size_t)N_ * C_ * KEPT_) * sizeof(float);
  if (ws_size < need) return;

  hipMemsetAsync(relsum, 0, P_ * sizeof(float), stream);

  // projections (q, k, v) -- fp32 WMMA GEMMs
  proj_gemm<<<dim3(256, 1, N_), 128, 0, stream>>>(
      Wq, bq, x, qm, CQK_, C_, P_, 1024, (long)N_ * C_ * 1024, 1024L,
      (long)C_ * 1024, (long)CQK_ * P_);
  proj_gemm<<<dim3(128, 1, N_), 128, 0, stream>>>(
      Wk, bk, f2, km, CQK_, C_, Q_, 4096, 0L, 4096L,
      (long)C_ * 4096, (long)CQK_ * Q_);
  proj_gemm<<<dim3(2048, 1, N_), 128, 0, stream>>>(
      Wv, bv, x, vm, C_, C_, P_, 1024, (long)N_ * C_ * 1024, 1024L,
      (long)C_ * 1024, (long)C_ * P_);

  // softmax stats + row importance (rel never materialized: 0.5 GB HBM saved)
  col_stats<<<dim3(Q_ / 16, 1, N_), 256, 0, stream>>>(qm, km, colmax, colden);
  rel_rowsum<<<dim3(P_ / 16 / 8, 1, N_), 256, 0, stream>>>(qm, km, colmax, colden, relsum);

  // top-50% selection + compaction
  select_topk<<<1, 256, 0, stream>>>(relsum, kept);
  compact_cols<<<dim3(KEPT_ / 256, N_ * CQK_), 256, 0, stream>>>(qm, qc, kept, P_);
  compact_cols<<<dim3(KEPT_ / 256, N_ * C_), 256, 0, stream>>>(vm, vc, kept, P_);

  // fused probability-regen + output GEMM
  out_gemm<<<dim3(Q_ / 16, 1, N_), 256, 0, stream>>>(qc, km, vc, colmax, colden, out);
}